// PoseNCP_54571854463100
// MI455X (gfx1250) — compile-verified
//
#include <hip/hip_runtime.h>

// Problem dims (from reference)
#define BATCH 64
#define SEQ 32
#define VF 512
#define IFD 256
#define FDIM 768      // VF + IFD
#define UNITS 768
#define UNF 6
#define EPSV 1e-8f
#define H1 128        // regressor hidden
#define OUT6 6
#define ROWS (BATCH * SEQ)   // 2048

typedef __attribute__((ext_vector_type(2))) float v2f;
typedef __attribute__((ext_vector_type(8))) float v8f;

__device__ __forceinline__ float fast_sigmoid(float z) {
  // sigmoid(z) = 1 / (1 + exp(-z)); exp(-z) = exp2(-z * log2(e))
  return __builtin_amdgcn_rcpf(1.0f + __builtin_amdgcn_exp2f(-1.44269504088896340736f * z));
}

// ---------------------------------------------------------------------------
// Kernel 1: LTC recurrence. One block per batch element, one thread per unit.
// Transcendental/VALU bound; parameters stream from L2 (19MB total, fits).
// ---------------------------------------------------------------------------
__global__ __launch_bounds__(UNITS) void ltc_kernel(
    const float* __restrict__ fv, const float* __restrict__ fi,
    const float* __restrict__ ts,
    const float* __restrict__ input_w, const float* __restrict__ input_b,
    const float* __restrict__ s_mu, const float* __restrict__ s_sigma,
    const float* __restrict__ s_w, const float* __restrict__ s_erev,
    const float* __restrict__ mu, const float* __restrict__ sigma,
    const float* __restrict__ w, const float* __restrict__ erev,
    const float* __restrict__ gleak, const float* __restrict__ vleak,
    const float* __restrict__ cm,
    const float* __restrict__ output_w, const float* __restrict__ output_b,
    float* __restrict__ hseq)   // (BATCH, SEQ, UNITS)
{
  __shared__ float xs[FDIM];
  __shared__ float vs[UNITS];
  const int b = blockIdx.x;
  const int u = threadIdx.x;

  const float gl  = gleak[u];
  const float glvl = gl * vleak[u];
  const float cmu = cm[u];
  const float ow  = output_w[u], ob = output_b[u];
  const float iw  = input_w[u],  ib = input_b[u];

  vs[u] = 0.0f;
  float vcur = 0.0f;
  __syncthreads();

  for (int t = 0; t < SEQ; ++t) {
    // stage fused input x = inp * input_w + input_b into LDS
    float raw = (u < VF) ? fv[(b * SEQ + t) * VF + u]
                         : fi[(b * SEQ + t) * IFD + (u - VF)];
    xs[u] = raw * iw + ib;
    const float elapsed = ts[b * (SEQ + 1) + t + 1] - ts[b * (SEQ + 1) + t];
    const float cmt = cmu * (float)UNF * __builtin_amdgcn_rcpf(elapsed);
    // warm L2 path for recurrent params used in the unfolds (global_prefetch_b8)
    __builtin_prefetch(&w[u], 0, 0);
    __builtin_prefetch(&erev[u], 0, 0);
    __syncthreads();

    // --- sensory phase: reduce over f for this unit's column ---
    float wns = 0.0f, wds = 0.0f;
    #pragma unroll 4
    for (int f = 0; f < FDIM; ++f) {
      const int idx = f * UNITS + u;               // coalesced across threads
      const float sact = s_w[idx] * fast_sigmoid(s_sigma[idx] * (xs[f] - s_mu[idx]));
      wns += sact * s_erev[idx];
      wds += sact;
    }

    // --- ODE unfolds ---
    for (int k = 0; k < UNF; ++k) {
      float num = wns, den = wds;
      #pragma unroll 4
      for (int uu = 0; uu < UNITS; ++uu) {
        const int idx = uu * UNITS + u;            // coalesced across threads
        const float a = w[idx] * fast_sigmoid(sigma[idx] * (vs[uu] - mu[idx]));
        num += a * erev[idx];
        den += a;
      }
      const float newv = (cmt * vcur + glvl + num) *
                         __builtin_amdgcn_rcpf(cmt + gl + den + EPSV);
      __syncthreads();          // all reads of vs done
      vs[u] = newv;
      vcur  = newv;
      __syncthreads();          // new state visible
    }

    hseq[(b * SEQ + t) * UNITS + u] = vcur * ow + ob;
  }
}

// ---------------------------------------------------------------------------
// Kernel 2: x = LeakyReLU(hseq @ W1 + b1) via V_WMMA_F32_16X16X4_F32.
// One wave per 16x16 output tile; K-loop over 768 in steps of 4.
// A layout: reg r, lane-half h -> K = r + 2h, M = lane%16.
// B layout: reg r, lane-half h -> K = r + 2h, N = lane%16.
// C/D layout: reg r, lane-half h -> M = r + 8h, N = lane%16.
// ---------------------------------------------------------------------------
__global__ __launch_bounds__(128) void gemm1_wmma(
    const float* __restrict__ hseq,   // ROWS x UNITS
    const float* __restrict__ W1,     // UNITS x H1
    const float* __restrict__ b1,     // H1
    float* __restrict__ xbuf)         // ROWS x H1
{
  const int lane = threadIdx.x & 31;
  const int wid  = blockIdx.x * 4 + (threadIdx.x >> 5);  // 0..1023
  const int mtile = wid >> 3;        // 0..127
  const int ntile = wid & 7;         // 0..7
  const int rc  = lane & 15;         // M-row for A, N-col for B/C
  const int kh  = lane >> 4;         // lane half
  const int M0 = mtile * 16, N0 = ntile * 16;

  const float* ap = hseq + (M0 + rc) * UNITS + 2 * kh;
  const float* bp = W1 + 2 * kh * H1 + N0 + rc;

  v8f c = {};
  for (int k0 = 0; k0 < UNITS; k0 += 4) {
    v2f a, bm;
    a.x  = ap[k0];
    a.y  = ap[k0 + 1];
    bm.x = bp[k0 * H1];
    bm.y = bp[(k0 + 1) * H1];
    c = __builtin_amdgcn_wmma_f32_16x16x4_f32(
        /*neg_a=*/false, a, /*neg_b=*/false, bm,
        /*c_mod=*/(short)0, c, /*reuse_a=*/false, /*reuse_b=*/false);
  }

  const float bias = b1[N0 + rc];
  #pragma unroll
  for (int r = 0; r < 8; ++r) {
    const int M = M0 + r + 8 * kh;
    float x = c[r] + bias;
    x = x > 0.0f ? x : 0.1f * x;                 // LeakyReLU(0.1)
    xbuf[M * H1 + N0 + rc] = x;
  }
}

// ---------------------------------------------------------------------------
// Kernel 3: pose = x @ W2 + b2  (2048 x 128 x 6, negligible cost)
// ---------------------------------------------------------------------------
__global__ void pose_kernel(const float* __restrict__ xbuf,
                            const float* __restrict__ W2,  // H1 x 6
                            const float* __restrict__ b2,  // 6
                            float* __restrict__ pose)      // ROWS x 6
{
  const int idx = blockIdx.x * blockDim.x + threadIdx.x;
  if (idx >= ROWS * OUT6) return;
  const int row = idx / OUT6, col = idx % OUT6;
  float acc = b2[col];
  const float* xp = xbuf + row * H1;
  #pragma unroll 8
  for (int k = 0; k < H1; ++k) acc += xp[k] * W2[k * OUT6 + col];
  pose[idx] = acc;
}

extern "C" void kernel_launch(void* const* d_in, const int* in_sizes, int n_in,
                              void* d_out, int out_size, void* d_ws, size_t ws_size,
                              hipStream_t stream) {
  (void)in_sizes; (void)n_in; (void)out_size; (void)ws_size;

  const float* fv       = (const float*)d_in[0];
  // d_in[1] fv_alter unused, d_in[3] dec unused (reference ignores them)
  const float* fi       = (const float*)d_in[2];
  const float* ts       = (const float*)d_in[4];
  const float* input_w  = (const float*)d_in[5];
  const float* input_b  = (const float*)d_in[6];
  const float* s_mu     = (const float*)d_in[7];
  const float* s_sigma  = (const float*)d_in[8];
  const float* s_w      = (const float*)d_in[9];
  const float* s_erev   = (const float*)d_in[10];
  const float* mu       = (const float*)d_in[11];
  const float* sigma    = (const float*)d_in[12];
  const float* w        = (const float*)d_in[13];
  const float* erev     = (const float*)d_in[14];
  const float* gleak    = (const float*)d_in[15];
  const float* vleak    = (const float*)d_in[16];
  const float* cm       = (const float*)d_in[17];
  const float* output_w = (const float*)d_in[18];
  const float* output_b = (const float*)d_in[19];
  const float* W1       = (const float*)d_in[20];
  const float* b1       = (const float*)d_in[21];
  const float* W2       = (const float*)d_in[22];
  const float* b2       = (const float*)d_in[23];

  float* pose = (float*)d_out;                    // ROWS*6 floats
  float* hseq = (float*)d_out + ROWS * OUT6;      // ROWS*UNITS floats
  float* xbuf = (float*)d_ws;                     // ROWS*H1 floats = 1 MB

  ltc_kernel<<<BATCH, UNITS, 0, stream>>>(
      fv, fi, ts, input_w, input_b, s_mu, s_sigma, s_w, s_erev,
      mu, sigma, w, erev, gleak, vleak, cm, output_w, output_b, hseq);

  gemm1_wmma<<<256, 128, 0, stream>>>(hseq, W1, b1, xbuf);

  pose_kernel<<<(ROWS * OUT6 + 255) / 256, 256, 0, stream>>>(xbuf, W2, b2, pose);
}